// LSTMTagger_65481071397343
// MI455X (gfx1250) — compile-verified
//
#include <hip/hip_runtime.h>
#include <hip/hip_bf16.h>
#include <math.h>

typedef __attribute__((ext_vector_type(2))) float v2f;
typedef __attribute__((ext_vector_type(8))) float v8f;

#define S_LEN 128
#define BATCH 128
#define EDIM  64
#define HDIM  8
#define NQG   8      // NQ
#define TDIM  12
#define FAN   72     // E + H
#define NST   256    // 2^H amplitudes
#define NSAMP (S_LEN * BATCH)   // 16384
#define NOPS  30

// ---- workspace layout (bytes) ----
static constexpr size_t OPS_OFF  = 0;                                   // 30*4 f32
static constexpr size_t FLAT_OFF = 1024;                                // NSAMP*HDIM f32
static constexpr size_t UR_OFF   = FLAT_OFF + (size_t)NSAMP * HDIM * 4; // 256*256 f32
static constexpr size_t UI_OFF   = UR_OFF + (size_t)NST * NST * 4;
static constexpr size_t PSI_OFF  = UI_OFF + (size_t)NST * NST * 4;      // NSAMP*256 f32
static constexpr size_t ZEXP_OFF = PSI_OFF + (size_t)NSAMP * NST * 4;   // NSAMP*8 f32

// ============================================================================
// Kernel 0: replicate np.random.RandomState(1234) op list (MT19937, legacy
// randint = masked-rejection on next64, uniform = rk_double). Single thread.
// ops[i] = {type(0=rx,1=ry,2=rz,3=cnot), wireA, wireB, angle}
// ============================================================================
__global__ void qrng_ops_kernel(float* __restrict__ ops) {
  if (threadIdx.x != 0 || blockIdx.x != 0) return;
  unsigned mt[624];
  int mti;
  mt[0] = 1234u;
  for (int i = 1; i < 624; ++i)
    mt[i] = 1812433253u * (mt[i - 1] ^ (mt[i - 1] >> 30)) + (unsigned)i;
  mti = 624;

  auto next32 = [&]() -> unsigned {
    if (mti >= 624) {
      for (int k = 0; k < 624; ++k) {
        unsigned y = (mt[k] & 0x80000000u) | (mt[(k + 1) % 624] & 0x7fffffffu);
        unsigned v = mt[(k + 397) % 624] ^ (y >> 1);
        if (y & 1u) v ^= 2567483615u;
        mt[k] = v;
      }
      mti = 0;
    }
    unsigned y = mt[mti++];
    y ^= y >> 11;
    y ^= (y << 7) & 2636928640u;
    y ^= (y << 15) & 4022730752u;
    y ^= y >> 18;
    return y;
  };
  auto next64 = [&]() -> unsigned long long {
    unsigned long long hi = next32();
    unsigned lo = next32();
    return (hi << 32) | (unsigned long long)lo;
  };
  auto bounded = [&](unsigned long long rng) -> unsigned { // inclusive max
    unsigned long long mask = rng;
    mask |= mask >> 1; mask |= mask >> 2; mask |= mask >> 4;
    mask |= mask >> 8; mask |= mask >> 16; mask |= mask >> 32;
    unsigned long long v;
    do { v = next64() & mask; } while (v > rng);
    return (unsigned)v;
  };
  auto uniform2pi = [&]() -> float {
    unsigned a = next32() >> 5, b = next32() >> 6;
    double d = (a * 67108864.0 + b) * (1.0 / 9007199254740992.0);
    return (float)(d * 6.283185307179586);
  };

  for (int i = 0; i < NOPS; ++i) {
    unsigned kind = bounded(3);
    float ty, wa = 0.f, wb = 0.f, ang = 0.f;
    if (kind == 3u) {
      unsigned c = bounded(7);   // randint(8)
      unsigned t = bounded(6);   // randint(7)
      if (t >= c) t += 1;
      ty = 3.0f; wa = (float)c; wb = (float)t;
    } else {
      ty = (float)(kind % 3u);   // 0 rx, 1 ry, 2 rz
      wa = (float)bounded(7);    // randint(8)
      ang = uniform2pi();
    }
    ops[i * 4 + 0] = ty; ops[i * 4 + 1] = wa;
    ops[i * 4 + 2] = wb; ops[i * 4 + 3] = ang;
  }
}

// ============================================================================
// Kernel 1: embedding + quantum-LSTM scan. grid = B/16 blocks of 512 threads,
// each block owns 16 independent batch rows; weights + state LDS-resident.
// Writes flat[(b*S + s)*H + h] (batch-major, as reference).
// ============================================================================
__global__ __launch_bounds__(512) void lstm_kernel(
    const float* __restrict__ emb,  const float* __restrict__ Win,
    const float* __restrict__ b_in, const float* __restrict__ phi,
    const float* __restrict__ Wout, const float* __restrict__ b_out,
    const int* __restrict__ sent,   float* __restrict__ flat)
{
  __shared__ float Wl[32 * FAN];          // Win (4*8, 72)
  __shared__ float binl[32], phil[32], boutl[32];
  __shared__ float Woutl[32 * 8];         // Wout (4*8, 8)
  __shared__ float comb[16][FAN];
  __shared__ float cz[16][32];
  __shared__ float hst[16][HDIM], cst[16][HDIM];

  const int tid = threadIdx.x;
  const int b0  = blockIdx.x * 16;

  for (int i = tid; i < 32 * FAN; i += 512) Wl[i] = Win[i];
  for (int i = tid; i < 256; i += 512) Woutl[i] = Wout[i];
  if (tid < 32) { binl[tid] = b_in[tid]; phil[tid] = phi[tid]; boutl[tid] = b_out[tid]; }
  if (tid < 128) { hst[tid >> 3][tid & 7] = 0.f; cst[tid >> 3][tid & 7] = 0.f; }
  __syncthreads();

  for (int s = 0; s < S_LEN; ++s) {
    // comb = [emb[token], h]
    for (int idx = tid; idx < 16 * FAN; idx += 512) {
      int b = idx / FAN, f = idx % FAN;
      float v;
      if (f < EDIM) {
        int tok = sent[s * BATCH + b0 + b];
        v = emb[(size_t)tok * EDIM + f];
      } else {
        v = hst[b][f - EDIM];
      }
      comb[b][f] = v;
    }
    __syncthreads();
    // z = comb @ Win^T + b_in + phi ; cz = cos(z)
    {
      int b = tid >> 5, gq = tid & 31;
      float acc = binl[gq] + phil[gq];
      const float* wr = &Wl[gq * FAN];
      #pragma unroll 8
      for (int f = 0; f < FAN; ++f) acc += comb[b][f] * wr[f];
      cz[b][gq] = cosf(acc);
    }
    __syncthreads();
    // cumprod over q within each gate
    if (tid < 64) {
      int b = tid >> 2, g = tid & 3;
      float r = 1.f;
      #pragma unroll
      for (int q = 0; q < NQG; ++q) { r *= cz[b][g * 8 + q]; cz[b][g * 8 + q] = r; }
    }
    __syncthreads();
    // gates + state update
    if (tid < 128) {
      int b = tid >> 3, h = tid & 7;
      float pre[4];
      #pragma unroll
      for (int g = 0; g < 4; ++g) {
        float a = boutl[g * 8 + h];
        #pragma unroll
        for (int q = 0; q < 8; ++q) a += cz[b][g * 8 + q] * Woutl[(g * 8 + h) * 8 + q];
        pre[g] = a;
      }
      float fg = 1.f / (1.f + expf(-pre[0]));
      float ig = 1.f / (1.f + expf(-pre[1]));
      float gg = tanhf(pre[2]);
      float og = 1.f / (1.f + expf(-pre[3]));
      float c  = fg * cst[b][h] + ig * gg;
      float hx = og * tanhf(c);
      cst[b][h] = c; hst[b][h] = hx;
      flat[((size_t)(b0 + b) * S_LEN + s) * HDIM + h] = hx;
    }
    __syncthreads();
  }
}

// ============================================================================
// Kernel 2: build fixed unitary U (256x256 complex) column-by-column: one
// block per column = one 8-qubit statevector sim of the 30 random ops.
// Wire w maps to bit (7-w) (axis-major flatten as in the reference).
// Writes row-major Ur[j][k], Ui[j][k].
// ============================================================================
__global__ __launch_bounds__(128) void buildU_kernel(
    const float* __restrict__ ops, float* __restrict__ Ur, float* __restrict__ Ui)
{
  __shared__ float sr[NST], si[NST];
  const int tid = threadIdx.x;
  const int col = blockIdx.x;
  for (int j = tid; j < NST; j += 128) { sr[j] = (j == col) ? 1.f : 0.f; si[j] = 0.f; }
  __syncthreads();

  for (int op = 0; op < NOPS; ++op) {
    int   ty  = (int)ops[op * 4 + 0];
    int   wa  = (int)ops[op * 4 + 1];
    int   wb  = (int)ops[op * 4 + 2];
    float ang = ops[op * 4 + 3];
    if (ty == 3) {                       // CNOT(wa -> wb)
      if (tid < 64) {
        int cm = 1 << (7 - wa), tm = 1 << (7 - wb);
        int lo = cm < tm ? cm : tm, hi = cm < tm ? tm : cm;
        int x = tid;
        x = ((x & ~(lo - 1)) << 1) | (x & (lo - 1));
        x = ((x & ~(hi - 1)) << 1) | (x & (hi - 1));
        int j0 = x | cm;                 // control=1, target=0
        int j1 = j0 | tm;
        float tr = sr[j0], ti = si[j0];
        sr[j0] = sr[j1]; si[j0] = si[j1];
        sr[j1] = tr;     si[j1] = ti;
      }
    } else {                             // single-qubit rotation, 128 pairs
      int mask = 1 << (7 - wa), low = mask - 1;
      float c = cosf(0.5f * ang), s = sinf(0.5f * ang);
      int j0 = ((tid & ~low) << 1) | (tid & low);
      int j1 = j0 | mask;
      float a0r = sr[j0], a0i = si[j0], a1r = sr[j1], a1i = si[j1];
      float n0r, n0i, n1r, n1i;
      if (ty == 0) {        // RX
        n0r = c * a0r + s * a1i;  n0i = c * a0i - s * a1r;
        n1r = c * a1r + s * a0i;  n1i = c * a1i - s * a0r;
      } else if (ty == 1) { // RY
        n0r = c * a0r - s * a1r;  n0i = c * a0i - s * a1i;
        n1r = s * a0r + c * a1r;  n1i = s * a0i + c * a1i;
      } else {              // RZ
        n0r = c * a0r + s * a0i;  n0i = c * a0i - s * a0r;
        n1r = c * a1r - s * a1i;  n1i = c * a1i + s * a1r;
      }
      sr[j0] = n0r; si[j0] = n0i; sr[j1] = n1r; si[j1] = n1i;
    }
    __syncthreads();
  }
  for (int j = tid; j < NST; j += 128) {
    Ur[(size_t)j * NST + col] = sr[j];
    Ui[(size_t)j * NST + col] = si[j];
  }
}

// ============================================================================
// Kernel 3: build real product state Psi[n][j] = prod_w (bit ? sin : cos)(h/2)
// ============================================================================
__global__ __launch_bounds__(256) void psi_kernel(
    const float* __restrict__ flat, float* __restrict__ Psi)
{
  __shared__ float cc[8], ss[8];
  const int n = blockIdx.x;
  const int j = threadIdx.x;
  if (j < 8) {
    float th = 0.5f * flat[(size_t)n * HDIM + j];
    cc[j] = cosf(th); ss[j] = sinf(th);
  }
  __syncthreads();
  float v = 1.f;
  #pragma unroll
  for (int w = 0; w < 8; ++w) v *= ((j >> (7 - w)) & 1) ? ss[w] : cc[w];
  Psi[(size_t)n * NST + j] = v;
}

// ============================================================================
// Kernel 4: Phi = Psi @ U^T (real & imag) via V_WMMA_F32_16X16X4_F32,
// then in-LDS RX(phiq) epilogue + Z measurement. One wave per 16 samples.
// ============================================================================
__global__ __launch_bounds__(32) void qhead_gemm_kernel(
    const float* __restrict__ Psi, const float* __restrict__ Ur,
    const float* __restrict__ Ui,  const float* __restrict__ phiq,
    float* __restrict__ zexp)
{
  __shared__ float psiL[16 * 257];   // padded rows: 16-bank-conflict free
  __shared__ float phr [16 * 257];
  __shared__ float phim[16 * 257];

  const int lane = threadIdx.x;
  const int m0   = blockIdx.x * 16;

  for (int idx = lane; idx < 16 * NST; idx += 32) {
    int m = idx >> 8, j = idx & 255;
    psiL[m * 257 + j] = Psi[(size_t)(m0 + m) * NST + j];
  }
  __syncthreads();

  const int mrow  = lane & 15;           // A: M row / B,D: N column offset
  const int khalf = (lane >> 4) << 1;    // K sub-offset 0 or 2
  const int mbase = (lane >> 4) << 3;    // D: M base 0 or 8

  for (int part = 0; part < 2; ++part) {
    const float* U     = part ? Ui : Ur;
    float*       stage = part ? phim : phr;
    for (int ntile = 0; ntile < 16; ++ntile) {
      const int n0 = ntile * 16;
      const int nn = n0 + mrow;
      v8f acc = {0.f, 0.f, 0.f, 0.f, 0.f, 0.f, 0.f, 0.f};
      #pragma unroll 4
      for (int k0 = 0; k0 < NST; k0 += 4) {
        int kk = k0 + khalf;
        v2f a; a.x = psiL[mrow * 257 + kk]; a.y = psiL[mrow * 257 + kk + 1];
        const float* up = &U[(size_t)nn * NST + kk];
        v2f b; b.x = up[0]; b.y = up[1];
        acc = __builtin_amdgcn_wmma_f32_16x16x4_f32(
            false, a, false, b, (short)0, acc, false, false);
      }
      #pragma unroll
      for (int v = 0; v < 8; ++v)
        stage[(mbase + v) * 257 + n0 + mrow] = acc[v];
    }
  }
  __syncthreads();

  // epilogue: RX(phiq[w]) layer + <Z_w> measurement, one lane per sample
  if (lane < 16) {
    float* rr = &phr [lane * 257];
    float* ii = &phim[lane * 257];
    for (int w = 0; w < 8; ++w) {
      float th = 0.5f * phiq[w];
      float c = cosf(th), s = sinf(th);
      int mask = 1 << (7 - w), low = mask - 1;
      for (int t = 0; t < 128; ++t) {
        int j0 = ((t & ~low) << 1) | (t & low);
        int j1 = j0 | mask;
        float a0r = rr[j0], a0i = ii[j0], a1r = rr[j1], a1i = ii[j1];
        float n0r = c * a0r + s * a1i, n0i = c * a0i - s * a1r;
        float n1r = c * a1r + s * a0i, n1i = c * a1i - s * a0r;
        rr[j0] = n0r; ii[j0] = n0i; rr[j1] = n1r; ii[j1] = n1i;
      }
    }
    float z[8] = {0.f, 0.f, 0.f, 0.f, 0.f, 0.f, 0.f, 0.f};
    for (int j = 0; j < NST; ++j) {
      float p = rr[j] * rr[j] + ii[j] * ii[j];
      #pragma unroll
      for (int w = 0; w < 8; ++w) z[w] += ((j >> (7 - w)) & 1) ? -p : p;
    }
    #pragma unroll
    for (int w = 0; w < 8; ++w) zexp[(size_t)(m0 + lane) * 8 + w] = z[w];
  }
}

// ============================================================================
// Kernel 5: logits = zexp @ Whead^T + b_head ; log_softmax over T=12.
// (reshape(S,B,T) is a pure reinterpretation: groups of 12 align with samples)
// ============================================================================
__global__ __launch_bounds__(256) void head_kernel(
    const float* __restrict__ zexp, const float* __restrict__ Whead,
    const float* __restrict__ bhead, float* __restrict__ out)
{
  int n = blockIdx.x * 256 + threadIdx.x;
  if (n >= NSAMP) return;
  float zv[8];
  #pragma unroll
  for (int h = 0; h < 8; ++h) zv[h] = zexp[(size_t)n * 8 + h];
  float l[TDIM], mx = -1e30f;
  #pragma unroll
  for (int t = 0; t < TDIM; ++t) {
    float a = bhead[t];
    #pragma unroll
    for (int h = 0; h < 8; ++h) a += zv[h] * Whead[t * 8 + h];
    l[t] = a; mx = fmaxf(mx, a);
  }
  float se = 0.f;
  #pragma unroll
  for (int t = 0; t < TDIM; ++t) se += expf(l[t] - mx);
  float lse = mx + logf(se);
  #pragma unroll
  for (int t = 0; t < TDIM; ++t) out[(size_t)n * TDIM + t] = l[t] - lse;
}

// ============================================================================
extern "C" void kernel_launch(void* const* d_in, const int* in_sizes, int n_in,
                              void* d_out, int out_size, void* d_ws, size_t ws_size,
                              hipStream_t stream)
{
  (void)in_sizes; (void)n_in; (void)out_size; (void)ws_size;
  const float* emb    = (const float*)d_in[0];
  const float* Win    = (const float*)d_in[1];
  const float* b_in   = (const float*)d_in[2];
  const float* phi    = (const float*)d_in[3];
  const float* Wout   = (const float*)d_in[4];
  const float* b_out  = (const float*)d_in[5];
  const float* phiq   = (const float*)d_in[6];
  const float* Whead  = (const float*)d_in[7];
  const float* b_head = (const float*)d_in[8];
  const int*   sent   = (const int*)d_in[9];
  float* out = (float*)d_out;

  char* ws   = (char*)d_ws;
  float* ops  = (float*)(ws + OPS_OFF);
  float* flat = (float*)(ws + FLAT_OFF);
  float* Ur   = (float*)(ws + UR_OFF);
  float* Ui   = (float*)(ws + UI_OFF);
  float* Psi  = (float*)(ws + PSI_OFF);
  float* zex  = (float*)(ws + ZEXP_OFF);

  qrng_ops_kernel<<<1, 32, 0, stream>>>(ops);
  lstm_kernel<<<BATCH / 16, 512, 0, stream>>>(emb, Win, b_in, phi, Wout, b_out, sent, flat);
  buildU_kernel<<<NST, 128, 0, stream>>>(ops, Ur, Ui);
  psi_kernel<<<NSAMP, 256, 0, stream>>>(flat, Psi);
  qhead_gemm_kernel<<<NSAMP / 16, 32, 0, stream>>>(Psi, Ur, Ui, phiq, zex);
  head_kernel<<<(NSAMP + 255) / 256, 256, 0, stream>>>(zex, Whead, b_head, out);
}